// DeformableConvolution_24361054503302
// MI455X (gfx1250) — compile-verified
//
#include <hip/hip_runtime.h>

// ---------------------------------------------------------------------------
// Deformable convolution, MI455X (gfx1250), fp32 throughout.
// Pipeline: zero/pack NHWC -> pack W pairs -> offset conv -> fused sample+WMMA
// Workspace layout (floats):
//   xnhwc : 8*66*66*64 = 2,230,272   (zero-padded NHWC input)
//   Wtp   : 576*64     =    36,864   (W_conv packed as [k/2][oc][k%2])
//   offT  : 32768*18   =   589,824   (per-pixel offsets, pixel-contiguous)
// Total ~11.4 MB — L2 resident.
// ---------------------------------------------------------------------------

typedef __attribute__((ext_vector_type(2))) float v2f;
typedef __attribute__((ext_vector_type(8))) float v8f;

#define NB   8
#define CI   64
#define CO   64
#define HH   64
#define WW   64
#define HP   66
#define WP   66

static constexpr int XNHWC_ELEMS = NB * HP * WP * CI;   // 2,230,272
static constexpr int WT_ELEMS    = 576 * CO;            // 36,864
static constexpr int OFFT_ELEMS  = NB * HH * WW * 18;   // 589,824

// ---- kernel 0: zero the padded NHWC buffer (border must be 0) --------------
__global__ void k_zero(float* __restrict__ p, int n) {
    for (int i = blockIdx.x * blockDim.x + threadIdx.x; i < n;
         i += gridDim.x * blockDim.x)
        p[i] = 0.f;
}

// ---- kernel 1: NCHW -> padded NHWC via LDS transpose -----------------------
__global__ __launch_bounds__(256) void k_pack_x(const float* __restrict__ x,
                                                float* __restrict__ xnhwc) {
    __shared__ float t[64 * 65];                 // stride 65: conflict-free
    const int bh = blockIdx.x;
    const int b  = bh >> 6;
    const int h  = bh & 63;
    for (int idx = threadIdx.x; idx < 4096; idx += 256) {
        const int c = idx >> 6, w = idx & 63;
        t[c * 65 + w] = x[((b * 64 + c) * 64 + h) * 64 + w];
    }
    __syncthreads();
    for (int idx = threadIdx.x; idx < 4096; idx += 256) {
        const int w = idx >> 6, c = idx & 63;
        xnhwc[((b * HP + (h + 1)) * WP + (w + 1)) * CI + c] = t[c * 65 + w];
    }
}

// ---- kernel 2: W_conv[oc][c][kh][kw] -> Wtp[kp][oc][r], k = 2*kp + r -------
// Pair-packed so a lane's two K-values (k0+kb, k0+kb+1) are one b64 load.
__global__ void k_pack_w(const float* __restrict__ Wc, float* __restrict__ Wtp) {
    const int e = blockIdx.x * blockDim.x + threadIdx.x;   // 36,864 threads
    if (e >= WT_ELEMS) return;
    const int kp = e >> 7;            // 0..287
    const int rr = e & 127;
    const int oc = rr >> 1;
    const int r  = rr & 1;
    const int k  = kp * 2 + r;        // j = n*64 + c
    const int c  = k & 63;
    const int n  = k >> 6;
    Wtp[e] = Wc[oc * 576 + c * 9 + n];                     // coalesced writes
}

// ---- kernel 3: offset conv (3x3, pad 1) -> offT[p][18] ---------------------
__global__ __launch_bounds__(64) void k_off_conv(const float* __restrict__ x,
                                                 const float* __restrict__ Woff,
                                                 const float* __restrict__ boff,
                                                 float* __restrict__ offT) {
    const int bh = blockIdx.x;
    const int b  = bh >> 6;
    const int h  = bh & 63;
    const int w  = threadIdx.x;
    float acc[18];
#pragma unroll
    for (int j = 0; j < 18; ++j) acc[j] = 0.f;

    for (int c = 0; c < CI; ++c) {
        const float* xc = x + (b * 64 + c) * 4096;
#pragma unroll
        for (int n = 0; n < 9; ++n) {
            const int hh = h + n / 3 - 1;
            const int ww = w + n % 3 - 1;
            const float xv = (hh >= 0 && hh < 64 && ww >= 0 && ww < 64)
                                 ? xc[hh * 64 + ww] : 0.f;
            const float* wo = Woff + c * 9 + n;
#pragma unroll
            for (int oc = 0; oc < 18; ++oc)
                acc[oc] = fmaf(xv, wo[oc * 576], acc[oc]);
        }
    }
    const int p = (b << 12) + (h << 6) + w;
#pragma unroll
    for (int oc = 0; oc < 18; ++oc) offT[p * 18 + oc] = acc[oc] + boff[oc];
}

// ---- kernel 4: fused bilinear sampling + f32 WMMA GEMM ---------------------
// Block = 128 threads (4 waves), owns 32 pixels x 64 output channels.
// Each wave: one 16-wide oc tile, TWO 16-pixel M-tiles sharing each B
// fragment -> two v_wmma_f32_16x16x4_f32 per b64 B-load (halves B traffic).
__global__ __launch_bounds__(128) void k_deform_wmma(
    const float* __restrict__ xnhwc, const float* __restrict__ Wtp,
    const float* __restrict__ offT, float* __restrict__ out) {
    __shared__ float sA[32 * 577];   // A tile, odd row stride (bank-friendly)
    __shared__ int   sQ[4 * 288];    // corner coords per (pixel, tap)
    __shared__ float sG[4 * 288];    // bilinear weights per (pixel, tap)

    const int tid   = threadIdx.x;
    const int pbase = blockIdx.x * 32;       // 32 consecutive pixels, same row
    const int b     = pbase >> 12;
    const int hw0   = pbase & 4095;
    const int h     = hw0 >> 6;
    const int w0    = hw0 & 63;

    // -- phase 0: coords + weights for 32 px * 9 taps -----------------------
    for (int idx = tid; idx < 288; idx += 128) {
        const int px = idx / 9;
        const int n  = idx - px * 9;
        const int p  = pbase + px;
        const float offx = offT[p * 18 + n];
        const float offy = offT[p * 18 + 9 + n];
        // padded coords: p0 = h+1 / w+1, tap offset = n/3-1 / n%3-1
        const float fx = (float)(h + n / 3) + offx;
        const float fy = (float)(w0 + px + (n % 3)) + offy;
        const float flx = floorf(fx), fly = floorf(fy);
        const int qx  = min(max((int)flx,     0), HP - 1);
        const int qy  = min(max((int)fly,     0), WP - 1);
        const int qx2 = min(max((int)flx + 1, 0), HP - 1);
        const int qy2 = min(max((int)fly + 1, 0), WP - 1);
        const float pxc = fminf(fmaxf(fx, 0.f), (float)(HP - 1));
        const float pyc = fminf(fmaxf(fy, 0.f), (float)(WP - 1));
        const float dxl = 1.f + ((float)qx  - pxc);
        const float dxh = 1.f - ((float)qx2 - pxc);
        const float dyl = 1.f + ((float)qy  - pyc);
        const float dyh = 1.f - ((float)qy2 - pyc);
        sQ[idx] = qx;  sQ[288 + idx] = qy;
        sQ[576 + idx] = qx2;  sQ[864 + idx] = qy2;
        sG[idx]       = dxl * dyl;   // lt
        sG[288 + idx] = dxh * dyh;   // rb
        sG[576 + idx] = dxl * dyh;   // lb
        sG[864 + idx] = dxh * dyl;   // rt
    }
    __syncthreads();

    // -- phase 1: build A tile (32 x 576), coalesced NHWC gathers -----------
    for (int e = tid; e < 32 * 576; e += 128) {
        const int px = e / 576;
        const int j  = e - px * 576;          // j = n*64 + c
        const int n  = j >> 6;
        const int c  = j & 63;
        const int q  = px * 9 + n;
        const int qx  = sQ[q],       qy  = sQ[288 + q];
        const int qx2 = sQ[576 + q], qy2 = sQ[864 + q];
        const float* xb = xnhwc + b * (HP * WP * CI) + c;
        const float v = sG[q]       * xb[(qx  * WP + qy ) * CI]
                      + sG[288 + q] * xb[(qx2 * WP + qy2) * CI]
                      + sG[576 + q] * xb[(qx  * WP + qy2) * CI]
                      + sG[864 + q] * xb[(qx2 * WP + qy ) * CI];
        sA[px * 577 + j] = v;
    }
    __syncthreads();

    // -- phase 2: per-wave 16-oc tile, two M-tiles, full-K WMMA -------------
    const int lane = tid & 31;
    const int wv   = tid >> 5;          // 0..3 -> oc tile
    const int oc0  = wv * 16;
    const int Mn   = lane & 15;         // M for A, N for B
    const int khalf = lane >> 4;        // lanes 0-15 -> K{0,1}, 16-31 -> K{2,3}
    const int kb    = khalf * 2;
    v8f acc0 = {0.f, 0.f, 0.f, 0.f, 0.f, 0.f, 0.f, 0.f};
    v8f acc1 = {0.f, 0.f, 0.f, 0.f, 0.f, 0.f, 0.f, 0.f};
    const float* sAr0 = sA + Mn * 577 + kb;          // pixels 0..15
    const float* sAr1 = sA + (16 + Mn) * 577 + kb;   // pixels 16..31
    // B pair base: Wtp[kp][oc][0..1], kp = k0/2 + khalf
    const float* Wb = Wtp + khalf * 128 + (oc0 + Mn) * 2;
    for (int k0 = 0; k0 < 576; k0 += 4) {
        const v2f bb = *(const v2f*)(Wb + (k0 >> 1) * 128);  // one b64 load
        v2f a0, a1;
        a0.x = sAr0[k0];  a0.y = sAr0[k0 + 1];
        a1.x = sAr1[k0];  a1.y = sAr1[k0 + 1];
        acc0 = __builtin_amdgcn_wmma_f32_16x16x4_f32(
            false, a0, false, bb, (short)0, acc0, false, false);
        acc1 = __builtin_amdgcn_wmma_f32_16x16x4_f32(
            false, a1, false, bb, (short)0, acc1, false, false);
    }
    __syncthreads();

    // -- phase 3: transpose C through LDS, coalesced NCHW stores ------------
    float* sC = sA;                    // reuse; stride 33 -> conflict-free
#pragma unroll
    for (int r = 0; r < 8; ++r) {
        const int M = r + (khalf << 3);           // pixel-in-M-tile
        sC[(oc0 + Mn) * 33 + M]      = acc0[r];
        sC[(oc0 + Mn) * 33 + 16 + M] = acc1[r];
    }
    __syncthreads();
    for (int e = tid; e < 2048; e += 128) {
        const int oc  = e >> 5;
        const int pxl = e & 31;
        out[((b * 64 + oc) << 12) + hw0 + pxl] = sC[oc * 33 + pxl];
    }
}

// ---------------------------------------------------------------------------
extern "C" void kernel_launch(void* const* d_in, const int* in_sizes, int n_in,
                              void* d_out, int out_size, void* d_ws, size_t ws_size,
                              hipStream_t stream) {
    const float* x    = (const float*)d_in[0];   // (8,64,64,64)
    const float* Woff = (const float*)d_in[1];   // (18,64,3,3)
    const float* boff = (const float*)d_in[2];   // (18,)
    const float* Wc   = (const float*)d_in[3];   // (64,64,3,3)
    float* out = (float*)d_out;                  // (8,64,64,64)

    float* wsf   = (float*)d_ws;
    float* xnhwc = wsf;
    float* Wtp   = xnhwc + XNHWC_ELEMS;
    float* offT  = Wtp + WT_ELEMS;

    k_zero<<<2048, 256, 0, stream>>>(xnhwc, XNHWC_ELEMS);
    k_pack_x<<<NB * HH, 256, 0, stream>>>(x, xnhwc);
    k_pack_w<<<(WT_ELEMS + 255) / 256, 256, 0, stream>>>(Wc, Wtp);
    k_off_conv<<<NB * HH, 64, 0, stream>>>(x, Woff, boff, offT);
    k_deform_wmma<<<(NB * HH * WW) / 32, 128, 0, stream>>>(xnhwc, Wtp, offT, out);
}